// Regressor_83983790506010
// MI455X (gfx1250) — compile-verified
//
#include <hip/hip_runtime.h>
#include <hip/hip_bf16.h>

// Problem constants (match reference)
#define B_  32
#define T_  4096
#define E_  256
#define H_  128
#define SEGS 32            // T-segments per batch for the streaming pass
#define WAVES_PER_BLOCK 8  // 256 threads / wave32

typedef __attribute__((ext_vector_type(2))) float v2f;
typedef __attribute__((ext_vector_type(8))) float v8f;

// ---------------------------------------------------------------------------
// u = Wv @ Wfc   (E x H) @ (H x 1) -> (E)
// ---------------------------------------------------------------------------
__global__ __launch_bounds__(256) void u_kernel(const float* __restrict__ Wv,
                                                const float* __restrict__ Wfc,
                                                float* __restrict__ u) {
    const int e = threadIdx.x;               // 256 threads == E
    float s = 0.f;
    const float* row = Wv + (size_t)e * H_;
    #pragma unroll 8
    for (int h = 0; h < H_; ++h) s = fmaf(row[h], Wfc[h], s);
    u[e] = s;
}

// ---------------------------------------------------------------------------
// M = Wq @ Wk^T  (E x E), K = H, fp32 WMMA 16x16x4.
// One wave computes one 16x16 tile; 32 WMMA steps over K=128.
// ISA layout (32-bit A 16x4): lane L holds rows M=L%16;
//   VGPR0 = K = 2*(L>=16), VGPR1 = K = 2*(L>=16)+1  -> contiguous float2.
// B = Wk^T has the mirrored layout: lane L holds cols N=L%16 of Wk rows.
// C/D: VGPR r -> row = r + 8*(L>=16), col = L%16.
// ---------------------------------------------------------------------------
__global__ __launch_bounds__(32) void m_wmma_kernel(const float* __restrict__ Wq,
                                                    const float* __restrict__ Wk,
                                                    float* __restrict__ M) {
    const int lane = threadIdx.x;            // 0..31 (wave32)
    const int i0 = blockIdx.y * 16;          // tile row in M
    const int j0 = blockIdx.x * 16;          // tile col in M
    const int m    = lane & 15;
    const int koff = (lane >> 4) << 1;       // 0 for lanes 0-15, 2 for 16-31

    const float* aptr = Wq + (size_t)(i0 + m) * H_ + koff;
    const float* bptr = Wk + (size_t)(j0 + m) * H_ + koff;

    v8f c = {};                              // zero accumulator
    #pragma unroll 4
    for (int h = 0; h < H_; h += 4) {
        v2f a = *(const v2f*)(aptr + h);     // {K=koff, K=koff+1}
        v2f b = *(const v2f*)(bptr + h);
        // 8 args: (neg_a, A, neg_b, B, c_mod, C, reuse_a, reuse_b)
        c = __builtin_amdgcn_wmma_f32_16x16x4_f32(
                false, a, false, b, (short)0, c, false, false);
    }

    const int row0 = i0 + ((lane >> 4) << 3);  // +8 for upper half-wave
    const int col  = j0 + m;
    #pragma unroll
    for (int r = 0; r < 8; ++r)
        M[(size_t)(row0 + r) * E_ + col] = c[r];
}

// ---------------------------------------------------------------------------
// Streaming pass over x (the only HBM-heavy kernel, ~134 MB read once):
// per row t: s = u . x_t ; y  += s * x_t ; xsum += x_t
// Block = (b, seg): 8 waves, each wave handles T/(SEGS*8) = 16 rows.
// Lane owns 8 contiguous floats of the 256-float row (two b128 loads).
// Deterministic: register accumulators -> LDS cross-wave reduce -> per-block
// partial slab; no floating-point atomics anywhere.
// ---------------------------------------------------------------------------
__global__ __launch_bounds__(256) void stream_kernel(const float* __restrict__ x,
                                                     const float* __restrict__ u,
                                                     float* __restrict__ part) {
    const int b    = blockIdx.y;
    const int seg  = blockIdx.x;
    const int w    = threadIdx.x >> 5;
    const int lane = threadIdx.x & 31;
    const int e0   = lane * 8;
    constexpr int ROWS = T_ / (SEGS * WAVES_PER_BLOCK);   // 16

    float uu[8];
    #pragma unroll
    for (int i = 0; i < 8; ++i) uu[i] = u[e0 + i];

    float accy[8] = {0.f,0.f,0.f,0.f,0.f,0.f,0.f,0.f};
    float accs[8] = {0.f,0.f,0.f,0.f,0.f,0.f,0.f,0.f};

    const int t0 = (seg * WAVES_PER_BLOCK + w) * ROWS;
    const float* xb = x + ((size_t)b * T_ + t0) * E_ + e0;

    for (int r = 0; r < ROWS; ++r) {
        const float4 p = *(const float4*)(xb);      // global_load_b128
        const float4 q = *(const float4*)(xb + 4);  // global_load_b128
        float s = fmaf(p.x, uu[0], fmaf(p.y, uu[1], fmaf(p.z, uu[2],
                  fmaf(p.w, uu[3], fmaf(q.x, uu[4], fmaf(q.y, uu[5],
                  fmaf(q.z, uu[6],       q.w * uu[7])))))));
        // wave32 butterfly reduce -> all lanes hold full dot u.x_t
        #pragma unroll
        for (int off = 16; off > 0; off >>= 1)
            s += __shfl_xor(s, off, 32);
        accy[0] = fmaf(s, p.x, accy[0]); accs[0] += p.x;
        accy[1] = fmaf(s, p.y, accy[1]); accs[1] += p.y;
        accy[2] = fmaf(s, p.z, accy[2]); accs[2] += p.z;
        accy[3] = fmaf(s, p.w, accy[3]); accs[3] += p.w;
        accy[4] = fmaf(s, q.x, accy[4]); accs[4] += q.x;
        accy[5] = fmaf(s, q.y, accy[5]); accs[5] += q.y;
        accy[6] = fmaf(s, q.z, accy[6]); accs[6] += q.z;
        accy[7] = fmaf(s, q.w, accy[7]); accs[7] += q.w;
        xb += E_;
    }

    // cross-wave reduce in LDS (16 KB)
    __shared__ float lds[WAVES_PER_BLOCK][2][E_];
    #pragma unroll
    for (int i = 0; i < 8; ++i) {
        lds[w][0][e0 + i] = accy[i];
        lds[w][1][e0 + i] = accs[i];
    }
    __syncthreads();

    const int e = threadIdx.x;               // 0..255
    float ty = 0.f, ts = 0.f;
    #pragma unroll
    for (int ww = 0; ww < WAVES_PER_BLOCK; ++ww) {
        ty += lds[ww][0][e];
        ts += lds[ww][1][e];
    }
    float* dst = part + (size_t)(b * SEGS + seg) * (2 * E_);
    dst[e]       = ty;   // y partial
    dst[E_ + e]  = ts;   // xsum partial
}

// ---------------------------------------------------------------------------
// Final: fold segment partials, then out[b] = scale * xbar^T M y + bfc.
// One block per batch.
// ---------------------------------------------------------------------------
__global__ __launch_bounds__(256) void final_kernel(const float* __restrict__ part,
                                                    const float* __restrict__ M,
                                                    const float* __restrict__ bfc,
                                                    float* __restrict__ out) {
    const int b = blockIdx.x;
    const int e = threadIdx.x;               // 0..255
    __shared__ float ysh[E_];
    __shared__ float xsh[E_];
    __shared__ float red[256];

    float ty = 0.f, ts = 0.f;
    const float* p = part + (size_t)b * SEGS * (2 * E_);
    for (int s = 0; s < SEGS; ++s) {
        ty += p[(size_t)s * (2 * E_) + e];
        ts += p[(size_t)s * (2 * E_) + E_ + e];
    }
    ysh[e] = ty;
    xsh[e] = ts * (1.0f / (float)T_);        // xbar
    __syncthreads();

    // thread e: (M y)[e], then weight by xbar[e]
    float t = 0.f;
    const float* Mrow = M + (size_t)e * E_;
    #pragma unroll 8
    for (int j = 0; j < E_; ++j) t = fmaf(Mrow[j], ysh[j], t);
    red[e] = xsh[e] * t;

    // block tree reduce
    for (int stp = 128; stp > 0; stp >>= 1) {
        __syncthreads();
        if (e < stp) red[e] += red[e + stp];
    }
    if (e == 0)
        out[b] = 0.0625f * red[0] + bfc[0];  // scale = E^-0.5 = 1/16
}

// ---------------------------------------------------------------------------
extern "C" void kernel_launch(void* const* d_in, const int* in_sizes, int n_in,
                              void* d_out, int out_size, void* d_ws, size_t ws_size,
                              hipStream_t stream) {
    const float* x   = (const float*)d_in[0];   // [B,T,E]
    const float* Wk  = (const float*)d_in[1];   // [E,H]
    const float* Wq  = (const float*)d_in[2];   // [E,H]
    const float* Wv  = (const float*)d_in[3];   // [E,H]
    const float* Wfc = (const float*)d_in[4];   // [H,1]
    const float* bfc = (const float*)d_in[5];   // [1]
    float* out = (float*)d_out;                 // [B,1]

    float* ws   = (float*)d_ws;
    float* u    = ws;                           // E floats
    float* M    = ws + E_;                      // E*E floats
    float* part = ws + E_ + (size_t)E_ * E_;    // B*SEGS*2*E floats

    u_kernel<<<1, 256, 0, stream>>>(Wv, Wfc, u);
    m_wmma_kernel<<<dim3(E_ / 16, E_ / 16), 32, 0, stream>>>(Wq, Wk, M);
    stream_kernel<<<dim3(SEGS, B_), 256, 0, stream>>>(x, u, part);
    final_kernel<<<B_, 256, 0, stream>>>(part, M, bfc, out);
}